// MyModel_17884243821103
// MI455X (gfx1250) — compile-verified
//
#include <hip/hip_runtime.h>
#include <hip/hip_bf16.h>

typedef __attribute__((ext_vector_type(2))) float v2f;
typedef __attribute__((ext_vector_type(8))) float v8f;

#define BLOCK_THREADS 256
#define KCHUNK 64

// ---------------------------------------------------------------------------
// GEMM: out[n x N] = act(A[n x K] @ W[K x N] + bias)
// One wave computes one 16x16 tile via V_WMMA_F32_16X16X4_F32, K-loop step 4.
// Block = 8 waves -> 16 rows x 128 cols. gridDim.x = ceil(n/16), gridDim.y = N/128.
// ACT: 0 = none, 1 = PReLU(a), 2 = ReLU
// ---------------------------------------------------------------------------
template <int K, int N, int ACT, bool HAS_BIAS>
__global__ __launch_bounds__(BLOCK_THREADS) void gemm_wmma_f32(
    const float* __restrict__ A, const float* __restrict__ W,
    const float* __restrict__ bias, const float* __restrict__ prelu_a,
    float* __restrict__ out, int n) {
  __shared__ __align__(16) float As[16 * 256];        // 16 rows x K (max 256)
  __shared__ __align__(16) float Wt[128 * KCHUNK];    // transposed W chunk: [col][k]

  const int tid = threadIdx.x;
  const int lane = tid & 31;
  const int wave = tid >> 5;
  const int rowBase = blockIdx.x * 16;
  const int colBase = blockIdx.y * 128;

  // Stage the 16-row A slab (full K) into LDS, coalesced float4 loads.
  const int kq = K >> 2;  // float4s per row
  for (int idx = tid; idx < 16 * kq; idx += BLOCK_THREADS) {
    const int r = idx / kq;
    const int k4 = idx - r * kq;
    const int row = rowBase + r;
    float4 v = make_float4(0.f, 0.f, 0.f, 0.f);
    if (row < n) v = ((const float4*)A)[(size_t)row * kq + k4];
    *(float4*)&As[r * K + k4 * 4] = v;
  }

  // WMMA operand addressing (ISA 16x4 A / 4x16 B layouts):
  //   lanes 0-15 : M/N = lane,     VGPR0/1 = K = k, k+1
  //   lanes 16-31: M/N = lane-16,  VGPR0/1 = K = k+2, k+3
  const int r = lane & 15;
  const int kh = (lane >> 4) << 1;              // 0 or 2
  const int cLocal = (wave << 4) + (lane & 15); // column within 128-wide block tile

  v8f acc = {0.f, 0.f, 0.f, 0.f, 0.f, 0.f, 0.f, 0.f};

  for (int kb = 0; kb < K; kb += KCHUNK) {
    __syncthreads();  // A slab ready / previous Wt chunk consumed
    // Stage transposed W chunk: Wt[c][kk] = W[kb+kk][colBase+c]
    for (int idx = tid; idx < 128 * KCHUNK; idx += BLOCK_THREADS) {
      const int c = idx & 127;
      const int kk = idx >> 7;
      Wt[c * KCHUNK + kk] = W[(size_t)(kb + kk) * N + colBase + c];
    }
    __syncthreads();

#pragma unroll
    for (int k = 0; k < KCHUNK; k += 4) {
      v2f a = *(const v2f*)&As[r * K + kb + k + kh];
      v2f b = *(const v2f*)&Wt[cLocal * KCHUNK + k + kh];
      acc = __builtin_amdgcn_wmma_f32_16x16x4_f32(
          /*neg_a=*/false, a, /*neg_b=*/false, b,
          /*c_mod=*/(short)0, acc, /*reuse_a=*/false, /*reuse_b=*/false);
    }
  }

  // Epilogue. C/D layout: VGPR j -> M = j (lanes 0-15) or 8+j (lanes 16-31), N = lane&15.
  const int col = colBase + cLocal;
  float bval = 0.f;
  if (HAS_BIAS) bval = bias[col];
  float aval = 0.f;
  if (ACT == 1) aval = *prelu_a;
  const int mBase = (lane >> 4) << 3;
#pragma unroll
  for (int j = 0; j < 8; ++j) {
    const int row = rowBase + mBase + j;
    if (row < n) {
      float v = acc[j];
      if (HAS_BIAS) v += bval;
      if (ACT == 1) v = (v >= 0.f) ? v : aval * v;
      if (ACT == 2) v = fmaxf(v, 0.f);
      out[(size_t)row * N + col] = v;
    }
  }
}

// ---------------------------------------------------------------------------
// Graph normalization + scatter kernels
// ---------------------------------------------------------------------------
__global__ void k_fill(float* __restrict__ p, float v, int n) {
  int i = blockIdx.x * blockDim.x + threadIdx.x;
  if (i < n) p[i] = v;
}

__global__ void k_deg(const int* __restrict__ dst, float* __restrict__ deg, int E) {
  int e = blockIdx.x * blockDim.x + threadIdx.x;
  if (e < E) atomicAdd(&deg[dst[e]], 1.0f);
}

__global__ void k_rsqrt_inplace(float* __restrict__ p, int n) {
  int i = blockIdx.x * blockDim.x + threadIdx.x;
  if (i < n) p[i] = 1.0f / sqrtf(p[i]);  // deg >= 1 (self-loop), matches rsqrt path
}

// out[i][:] = dinv[i]^2 * h[i][:]   (self-loop contribution; also initializes out)
template <int C>
__global__ void k_self_init(float* __restrict__ out, const float* __restrict__ h,
                            const float* __restrict__ dinv, int n) {
  const int cq = C >> 2;
  int i = blockIdx.x * blockDim.x + threadIdx.x;  // float4 index
  if (i >= n * cq) return;
  const int node = i / cq;
  float s = dinv[node];
  s = s * s;
  float4 v = ((const float4*)h)[i];
  ((float4*)out)[i] = make_float4(v.x * s, v.y * s, v.z * s, v.w * s);
}

// per-edge: out[dst][:] += dinv[src]*dinv[dst] * h[src][:]   (C/4 threads per edge)
template <int C>
__global__ void k_edge_scatter(const int* __restrict__ src, const int* __restrict__ dst,
                               const float* __restrict__ dinv, const float* __restrict__ h,
                               float* __restrict__ out, int E) {
  const int tpe = C >> 2;
  const long long gid = (long long)blockIdx.x * blockDim.x + threadIdx.x;
  const long long e = gid / tpe;
  if (e >= E) return;
  const int c4 = (int)(gid - e * tpe);
  const int s = src[e];
  const int d = dst[e];
  const float norm = dinv[s] * dinv[d];
  float4 v = ((const float4*)h)[(long long)s * tpe + c4];
  float* o = out + (long long)d * C + c4 * 4;
  atomicAdd(o + 0, v.x * norm);
  atomicAdd(o + 1, v.y * norm);
  atomicAdd(o + 2, v.z * norm);
  atomicAdd(o + 3, v.w * norm);
}

// x = prelu(x + bias[c], a), elementwise in place
template <int C>
__global__ void k_bias_prelu(float* __restrict__ x, const float* __restrict__ bias,
                             const float* __restrict__ pa, long long total) {
  long long i = (long long)blockIdx.x * blockDim.x + threadIdx.x;
  if (i >= total) return;
  const int c = (int)(i & (C - 1));
  float v = x[i] + bias[c];
  const float a = *pa;
  x[i] = (v >= 0.f) ? v : a * v;
}

// ---------------------------------------------------------------------------
// Driver
// ---------------------------------------------------------------------------
extern "C" void kernel_launch(void* const* d_in, const int* in_sizes, int n_in,
                              void* d_out, int out_size, void* d_ws, size_t ws_size,
                              hipStream_t stream) {
  const float* x    = (const float*)d_in[0];
  const int*   ei   = (const int*)d_in[1];
  const float* W1   = (const float*)d_in[2];
  const float* b1   = (const float*)d_in[3];
  const float* W2   = (const float*)d_in[4];
  const float* b2   = (const float*)d_in[5];
  const float* pa   = (const float*)d_in[6];
  const float* fc1w = (const float*)d_in[7];
  const float* fc1b = (const float*)d_in[8];
  const float* fc2w = (const float*)d_in[9];
  const float* fc2b = (const float*)d_in[10];

  const int DIN = 128, DHID = 256, DOUT = 128;
  const int n = in_sizes[0] / DIN;
  const int E = in_sizes[1] / 2;
  const int* srcIdx = ei;
  const int* dstIdx = ei + E;

  float* ws = (float*)d_ws;
  float* dinv = ws;                                   // n floats (deg -> dinv)
  size_t off = ((size_t)n + 127) & ~(size_t)127;
  float* buf1 = ws + off;                             // n*256 floats
  float* buf2 = buf1 + (size_t)n * DHID;              // n*256 floats
  float* outNode = (float*)d_out;                     // n*128 (tuple element 0)
  float* outProj = outNode + (size_t)n * DOUT;        // n*128 (tuple element 1)

  const int B = 256;
  // 1) degrees (with self-loop) -> dinv = 1/sqrt(deg)
  k_fill<<<(n + B - 1) / B, B, 0, stream>>>(dinv, 1.0f, n);
  k_deg<<<(E + B - 1) / B, B, 0, stream>>>(dstIdx, dinv, E);
  k_rsqrt_inplace<<<(n + B - 1) / B, B, 0, stream>>>(dinv, n);

  const dim3 gRows16((n + 15) / 16, 1);

  // 2) h1 = x @ W1  -> buf1   [n,128]x[128,256]
  gemm_wmma_f32<128, 256, 0, false>
      <<<dim3((n + 15) / 16, 2), B, 0, stream>>>(x, W1, nullptr, nullptr, buf1, n);

  // 3) out1 = A_hat @ h1 -> buf2 ; then += b1, PReLU
  {
    int q = n * (DHID / 4);
    k_self_init<DHID><<<(q + B - 1) / B, B, 0, stream>>>(buf2, buf1, dinv, n);
    long long t = (long long)E * (DHID / 4);
    k_edge_scatter<DHID><<<(unsigned)((t + B - 1) / B), B, 0, stream>>>(
        srcIdx, dstIdx, dinv, buf1, buf2, E);
    long long tot = (long long)n * DHID;
    k_bias_prelu<DHID><<<(unsigned)((tot + B - 1) / B), B, 0, stream>>>(buf2, b1, pa, tot);
  }

  // 4) h2 = out1 @ W2 -> buf1   [n,256]x[256,128]
  gemm_wmma_f32<256, 128, 0, false>
      <<<gRows16, B, 0, stream>>>(buf2, W2, nullptr, nullptr, buf1, n);

  // 5) out2 = A_hat @ h2 -> d_out[0:n*128] ; then += b2, PReLU  (tuple element 0)
  {
    int q = n * (DOUT / 4);
    k_self_init<DOUT><<<(q + B - 1) / B, B, 0, stream>>>(outNode, buf1, dinv, n);
    long long t = (long long)E * (DOUT / 4);
    k_edge_scatter<DOUT><<<(unsigned)((t + B - 1) / B), B, 0, stream>>>(
        srcIdx, dstIdx, dinv, buf1, outNode, E);
    long long tot = (long long)n * DOUT;
    k_bias_prelu<DOUT><<<(unsigned)((tot + B - 1) / B), B, 0, stream>>>(outNode, b2, pa, tot);
  }

  // 6) proj1 = relu(out2 @ fc1w + fc1b) -> buf1
  gemm_wmma_f32<128, 128, 2, true>
      <<<gRows16, B, 0, stream>>>(outNode, fc1w, fc1b, nullptr, buf1, n);

  // 7) proj = proj1 @ fc2w + fc2b -> d_out[n*128:]  (tuple element 1)
  gemm_wmma_f32<128, 128, 0, true>
      <<<gRows16, B, 0, stream>>>(buf1, fc2w, fc2b, nullptr, outProj, n);
}